// ParametrisedPooling_67070209294882
// MI455X (gfx1250) — compile-verified
//
#include <hip/hip_runtime.h>
#include <stdint.h>

// ParametrisedPooling: out[i] = sum_k w[t,k] * x_in[idx[t,k], :],  t = indices_target[i]
// Fuses gather + weighted sum + output permutation. Bandwidth-bound (~600 MB total
// traffic -> ~25us at 23.3 TB/s). Uses gfx1250 async global->LDS gather path.

#define FD     128          // feature dim
#define N6C    163830
#define N7C    12
#define NOUT   (N6C + N7C)  // 163842
#define WPB    8            // waves per block (wave32)
#define TPB    (WPB * 32)

__global__ __launch_bounds__(TPB) void pool67_kernel(
    const float* __restrict__ x,
    const float* __restrict__ w6,
    const float* __restrict__ w7,
    const int*   __restrict__ i6,
    const int*   __restrict__ i7,
    const int*   __restrict__ tgt,
    float*       __restrict__ out)
{
    __shared__ float smem[WPB][7 * FD];   // 7 staged neighbor rows per wave

    const int lane = threadIdx.x & 31;
    const int wave = threadIdx.x >> 5;
    const int row  = blockIdx.x * WPB + wave;
    if (row >= NOUT) return;

    const int t = tgt[row];               // wave-uniform
    const int*   idx;
    const float* w;
    int K;
    if (t < N6C) { idx = i6 + t * 6;                w = w6 + t * 6;  K = 6; }
    else         { const int u = t - N6C;
                   idx = i7 + u * 7;                w = w7 + u * 7;  K = 7; }

    // Pull neighbor indices + weights into registers (broadcast loads, uniform addr).
    int   nk[7];
    float wk[7];
#pragma unroll
    for (int k = 0; k < 7; ++k) {
        nk[k] = (k < K) ? idx[k] : 0;
        wk[k] = (k < K) ? w[k]   : 0.0f;
    }

    // Low 32 bits of a generic LDS pointer == workgroup-relative LDS byte offset
    // that is what the async
    // instruction's LDS-address VGPR expects.
    const unsigned lbase =
        (unsigned)(uintptr_t)(&smem[wave][0]) + (unsigned)(lane * 16);

    // Issue all K gathers: each lane moves 16 B, a wave moves one 512 B row.
    // Tracked on ASYNCcnt; they pipeline without consuming VGPR return slots.
#pragma unroll
    for (int k = 0; k < 7; ++k) {
        if (k < K) {
            const float* gsrc = x + (size_t)nk[k] * FD + (size_t)(lane * 4);
            const unsigned laddr = lbase + (unsigned)(k * (FD * 4));
            asm volatile("global_load_async_to_lds_b128 %0, %1, off"
                         :: "v"(laddr), "v"(gsrc)
                         : "memory");
        }
    }

    // Wait for all async LDS writes of this wave; memory clobber = compiler barrier
    // so the LDS reads below cannot be hoisted above the wait.
    asm volatile("s_wait_asynccnt 0x0" ::: "memory");

    float4 acc = make_float4(0.f, 0.f, 0.f, 0.f);
    const float4* srow = (const float4*)(&smem[wave][0]);
#pragma unroll
    for (int k = 0; k < 7; ++k) {
        if (k < K) {
            const float4 v = srow[k * (FD / 4) + lane];
            acc.x = fmaf(wk[k], v.x, acc.x);
            acc.y = fmaf(wk[k], v.y, acc.y);
            acc.z = fmaf(wk[k], v.z, acc.z);
            acc.w = fmaf(wk[k], v.w, acc.w);
        }
    }

    float4* orow = (float4*)(out + (size_t)row * FD);
    orow[lane] = acc;   // coalesced 512 B store per wave
}

extern "C" void kernel_launch(void* const* d_in, const int* in_sizes, int n_in,
                              void* d_out, int out_size, void* d_ws, size_t ws_size,
                              hipStream_t stream) {
    (void)in_sizes; (void)n_in; (void)out_size; (void)d_ws; (void)ws_size;
    const float* x   = (const float*)d_in[0];   // x_in       [N_SRC,128] f32
    const float* w6  = (const float*)d_in[1];   // weights_6  [N6,6] f32
    const float* w7  = (const float*)d_in[2];   // weights_7  [N7,7] f32
    const int*   i6  = (const int*)  d_in[3];   // indices_6  [N6,6] i32
    const int*   i7  = (const int*)  d_in[4];   // indices_7  [N7,7] i32
    const int*   tgt = (const int*)  d_in[5];   // indices_target [N6+N7] i32
    float* out = (float*)d_out;                 // [N6+N7,128] f32

    const int blocks = (NOUT + WPB - 1) / WPB;  // 20481
    pool67_kernel<<<blocks, TPB, 0, stream>>>(x, w6, w7, i6, i7, tgt, out);
}